// TextSpace_85306640433864
// MI455X (gfx1250) — compile-verified
//
#include <hip/hip_runtime.h>
#include <math.h>

typedef float v2f __attribute__((ext_vector_type(2)));
typedef float v8f __attribute__((ext_vector_type(8)));

#define LVAL     4
#define NH       30
#define NPLANE   31
#define S_TOT    4096
#define D_TOT    768
#define CELLS    81
#define KPAD     96
#define ERSTRIDE (NPLANE * D_TOT)   // stride between grid cells in embeddings
#define CH       64                 // scan chunk size
#define NCH      (S_TOT / CH)       // 64 chunks

// ---------------- zero the dense bilinear-coefficient matrix ----------------
__global__ void k_zero(float* __restrict__ p, int n) {
    int i = blockIdx.x * blockDim.x + threadIdx.x;
    if (i < n) p[i] = 0.0f;
}

// ---- per-sample prep: bilinear coeffs -> dense A row (4 nonzeros), dist ----
__global__ void k_prep(const float* __restrict__ samples,
                       const float* __restrict__ last_point,
                       float* __restrict__ Adense,
                       float* __restrict__ dist) {
    int s = blockIdx.x * blockDim.x + threadIdx.x;
    if (s >= S_TOT) return;
    float x = samples[2 * s], y = samples[2 * s + 1];
    float nx, ny;
    if (s + 1 < S_TOT) { nx = samples[2 * s + 2]; ny = samples[2 * s + 3]; }
    else               { nx = last_point[0];      ny = last_point[1]; }
    float ex = nx - x, ey = ny - y;
    dist[s] = sqrtf(ex * ex + ey * ey);

    float dx = x - floorf(x);
    float dy = y - floorf(y);
    float xs = x + (float)LVAL, ys = y + (float)LVAL;
    int fx = (int)floorf(xs), cx = (int)ceilf(xs);
    int fy = (int)floorf(ys), cy = (int)ceilf(ys);

    float* row = Adense + (size_t)s * KPAD;
    // += (not =) so degenerate corners (dx==0 or dy==0 => duplicate cells) sum
    row[fx * 9 + fy] += (1.0f - dx) * (1.0f - dy);
    row[cx * 9 + fy] += dx * (1.0f - dy);
    row[fx * 9 + cy] += (1.0f - dx) * dy;
    row[cx * 9 + cy] += dx * dy;
}

// ------- GEMM1: dd = relu(A @ E_row30) * dist  via V_WMMA_F32_16X16X4 -------
__global__ void k_gemm_density(const float* __restrict__ Adense,
                               const float* __restrict__ emb,
                               const float* __restrict__ dist,
                               float* __restrict__ dd) {
    const int lane = threadIdx.x;        // 0..31, wave32
    const int idx  = lane & 15;
    const int half = lane >> 4;
    const int dBase = blockIdx.x * 16;                               // N tile
    const int sBase = (blockIdx.y * blockDim.y + threadIdx.y) * 16;  // M tile
    const float* Erow = emb + (size_t)NH * D_TOT;   // + cell*ERSTRIDE

    v8f acc = {};
    for (int kb = 0; kb < KPAD; kb += 4) {
        int k0 = kb + 2 * half;
        v2f a, b;
        a.x = Adense[(size_t)(sBase + idx) * KPAD + k0];
        a.y = Adense[(size_t)(sBase + idx) * KPAD + k0 + 1];
        // clamp padded K rows (A columns 81..95 are zero, product is exact 0)
        int r0 = (k0     < CELLS) ? k0       : 0;
        int r1 = (k0 + 1 < CELLS) ? (k0 + 1) : 0;
        b.x = Erow[(size_t)r0 * ERSTRIDE + dBase + idx];
        b.y = Erow[(size_t)r1 * ERSTRIDE + dBase + idx];
        acc = __builtin_amdgcn_wmma_f32_16x16x4_f32(
            false, a, false, b, (short)0, acc, false, false);
    }
#pragma unroll
    for (int r = 0; r < 8; ++r) {
        int s = sBase + r + 8 * half;
        float v = acc[r];
        v = v > 0.0f ? v : 0.0f;                 // relu
        dd[(size_t)s * D_TOT + dBase + idx] = v * dist[s];
    }
}

// ----------------- chunked exclusive scan of dd over samples ----------------
__global__ void k_chunksum(const float* __restrict__ dd, float* __restrict__ csum) {
    int t = blockIdx.x * blockDim.x + threadIdx.x;   // NCH*D_TOT threads
    if (t >= NCH * D_TOT) return;
    int c = t / D_TOT, d = t - c * D_TOT;
    float s = 0.0f;
    for (int i = 0; i < CH; ++i) s += dd[(size_t)(c * CH + i) * D_TOT + d];
    csum[t] = s;
}

__global__ void k_chunkscan(float* __restrict__ csum) {  // exclusive, in place
    int d = blockIdx.x * blockDim.x + threadIdx.x;
    if (d >= D_TOT) return;
    float cum = 0.0f;
    for (int c = 0; c < NCH; ++c) {
        size_t o = (size_t)c * D_TOT + d;
        float v = csum[o];
        csum[o] = cum;
        cum += v;
    }
}

// w[s,d] = exp(-cum_excl) * (1 - exp(-dd)) ; overwrites dd in place
__global__ void k_weights(float* __restrict__ ddw, const float* __restrict__ csum) {
    int t = blockIdx.x * blockDim.x + threadIdx.x;   // NCH*D_TOT threads
    if (t >= NCH * D_TOT) return;
    int c = t / D_TOT, d = t - c * D_TOT;
    float cum = csum[t];
    for (int i = 0; i < CH; ++i) {
        size_t o = (size_t)(c * CH + i) * D_TOT + d;
        float v = ddw[o];
        ddw[o] = __expf(-cum) * (1.0f - __expf(-v));
        cum += v;
    }
}

// --------- GEMM2: W2 = A^T @ w  (M=96, N=768, K=4096) via f32 WMMA ----------
__global__ void k_gemm_w2(const float* __restrict__ Adense,
                          const float* __restrict__ w,
                          float* __restrict__ W2) {
    const int lane = threadIdx.x;
    const int idx  = lane & 15;
    const int half = lane >> 4;
    const int dBase = blockIdx.x * 16;
    const int mBase = (blockIdx.y * blockDim.y + threadIdx.y) * 16;

    v8f acc = {};
    for (int kb = 0; kb < S_TOT; kb += 4) {
        int k0 = kb + 2 * half;
        v2f a, b;
        a.x = Adense[(size_t)k0 * KPAD + mBase + idx];        // A^T[m][k] = A[k][m]
        a.y = Adense[(size_t)(k0 + 1) * KPAD + mBase + idx];
        b.x = w[(size_t)k0 * D_TOT + dBase + idx];
        b.y = w[(size_t)(k0 + 1) * D_TOT + dBase + idx];
        acc = __builtin_amdgcn_wmma_f32_16x16x4_f32(
            false, a, false, b, (short)0, acc, false, false);
    }
#pragma unroll
    for (int r = 0; r < 8; ++r) {
        int m = mBase + r + 8 * half;
        W2[(size_t)m * D_TOT + dBase + idx] = acc[r];
    }
}

// -------- out[n,d] = sum_c W2[c,d] * E[c,n,d]  (81-term reduction) ----------
__global__ void k_combine(const float* __restrict__ W2,
                          const float* __restrict__ emb,
                          float* __restrict__ out) {
    int t = blockIdx.x * blockDim.x + threadIdx.x;
    if (t >= NH * D_TOT) return;
    int n = t / D_TOT, d = t - n * D_TOT;
    float acc = 0.0f;
    for (int c = 0; c < CELLS; ++c)
        acc = fmaf(W2[(size_t)c * D_TOT + d],
                   emb[((size_t)c * NPLANE + n) * D_TOT + d], acc);
    out[t] = acc;
}

extern "C" void kernel_launch(void* const* d_in, const int* in_sizes, int n_in,
                              void* d_out, int out_size, void* d_ws, size_t ws_size,
                              hipStream_t stream) {
    (void)in_sizes; (void)n_in; (void)out_size; (void)ws_size;
    const float* samples    = (const float*)d_in[0];
    const float* last_point = (const float*)d_in[1];
    const float* emb        = (const float*)d_in[2];
    float* out = (float*)d_out;

    float* ws     = (float*)d_ws;
    float* Adense = ws;                                   // 4096*96
    float* dist   = Adense + (size_t)S_TOT * KPAD;        // 4096
    float* ddw    = dist + S_TOT;                         // 4096*768 (dd, then w)
    float* W2     = ddw + (size_t)S_TOT * D_TOT;          // 96*768
    float* csum   = W2 + (size_t)KPAD * D_TOT;            // 64*768

    k_zero<<<(S_TOT * KPAD + 255) / 256, 256, 0, stream>>>(Adense, S_TOT * KPAD);
    k_prep<<<(S_TOT + 255) / 256, 256, 0, stream>>>(samples, last_point, Adense, dist);

    dim3 b1(32, 4), g1(D_TOT / 16, S_TOT / 16 / 4);       // 48 x 64 blocks
    k_gemm_density<<<g1, b1, 0, stream>>>(Adense, emb, dist, ddw);

    k_chunksum<<<(NCH * D_TOT + 255) / 256, 256, 0, stream>>>(ddw, csum);
    k_chunkscan<<<(D_TOT + 255) / 256, 256, 0, stream>>>(csum);
    k_weights<<<(NCH * D_TOT + 255) / 256, 256, 0, stream>>>(ddw, csum);

    dim3 b2(32, 2), g2(D_TOT / 16, KPAD / 16 / 2);        // 48 x 3 blocks
    k_gemm_w2<<<g2, b2, 0, stream>>>(Adense, ddw, W2);

    k_combine<<<(NH * D_TOT + 255) / 256, 256, 0, stream>>>(W2, emb, out);
}